// Moma_12206297055645
// MI455X (gfx1250) — compile-verified
//
#include <hip/hip_runtime.h>
#include <hip/hip_bf16.h>
#include <math.h>

// Problem constants (from reference)
#define B_     128
#define M_     384     // modules
#define FEAT_  1536    // 4*M
#define CLS_   1024
#define NG_    20000   // genes per modality

typedef float v2f __attribute__((ext_vector_type(2)));
typedef float v8f __attribute__((ext_vector_type(8)));

// Up to 6 independent (A, W, bias, C) GEMM problems sharing one launch.
struct GemmBatch {
    const float* A[6];
    const float* W[6];
    const float* bias[6];
    float*       C[6];
};

// C[M x Nout] = A[M x K] @ W[Nout x K]^T  (+bias, +relu), fp32 in/out.
// One wave per 16x16 output tile, K-loop with 4x v_wmma_f32_16x16x4_f32 per iter.
// A 16x4 tile layout: lane&15 = M row; lanes 0-15 hold K=0,1; lanes 16-31 hold K=2,3.
// B 4x16 tile layout: lane&15 = N col; same K split. Both are K-contiguous -> float2 loads.
__global__ __launch_bounds__(32)
void wmma_gemm_f32(GemmBatch gb, int K, int Nout, int relu) {
    const int g = blockIdx.z;
    const float* __restrict__ A = gb.A[g];
    const float* __restrict__ W = gb.W[g];
    const float* bias           = gb.bias[g];
    float* __restrict__ C       = gb.C[g];

    const int lane = threadIdx.x;          // 0..31 (wave32)
    const int half = lane >> 4;            // 0: K=0,1  1: K=2,3
    const int lid  = lane & 15;
    const int koff = half * 2;
    const int mt   = blockIdx.x * 16;      // batch tile
    const int nt   = blockIdx.y * 16;      // output-feature tile

    const float* Ap = A + (size_t)(mt + lid) * K + koff;
    const float* Wp = W + (size_t)(nt + lid) * K + koff;

    v8f acc = {};
    for (int kb = 0; kb < K; kb += 16) {
#pragma unroll
        for (int u = 0; u < 4; ++u) {
            v2f a = *reinterpret_cast<const v2f*>(Ap + kb + 4 * u);
            v2f b = *reinterpret_cast<const v2f*>(Wp + kb + 4 * u);
            // (neg_a, A, neg_b, B, c_mod, C, reuse_a, reuse_b)
            acc = __builtin_amdgcn_wmma_f32_16x16x4_f32(
                false, a, false, b, (short)0, acc, false, false);
        }
    }

    const float bv = bias ? bias[nt + lid] : 0.0f;
#pragma unroll
    for (int r = 0; r < 8; ++r) {
        const int row = mt + r + half * 8;
        float v = acc[r] + bv;
        if (relu) v = fmaxf(v, 0.0f);
        C[(size_t)row * Nout + nt + lid] = v;
    }
}

// Fused: pair L2-normalization + all six rank-2 softmax-attention bmms.
// One block per batch sample; the six 384-float unit vectors live in LDS.
// Energy matrices are rank-2 (unit 2-vectors => logits in [-1,1], exp is safe).
__global__ __launch_bounds__(M_)
void attn_kernel(const float* __restrict__ fc1,  // 6 x [B_ x M_]: ex,ey,mx,my,cx,cy raw
                 float* __restrict__ ef, float* __restrict__ mf, float* __restrict__ cf) {
    __shared__ float ex[M_], ey[M_], mx[M_], my[M_], cx[M_], cy[M_], zmc[M_];
    const int b = blockIdx.x;
    const int n = threadIdx.x;
    const size_t st  = (size_t)B_ * M_;
    const size_t off = (size_t)b * M_ + n;

    {
        float vx = fc1[0 * st + off], vy = fc1[1 * st + off];
        float inv = rsqrtf(vx * vx + vy * vy);
        ex[n] = vx * inv; ey[n] = vy * inv;
        vx = fc1[2 * st + off]; vy = fc1[3 * st + off];
        inv = rsqrtf(vx * vx + vy * vy);
        mx[n] = vx * inv; my[n] = vy * inv;
        vx = fc1[4 * st + off]; vy = fc1[5 * st + off];
        inv = rsqrtf(vx * vx + vy * vy);
        cx[n] = vx * inv; cy[n] = vy * inv;
    }
    __syncthreads();

    // Row sums of exp(E_mc) (A_mc normalizes over the NON-contracted axis).
    {
        const float q0 = mx[n], q1 = my[n];
        float z = 0.0f;
        for (int i = 0; i < M_; ++i) z += __expf(fmaf(q0, cx[i], q1 * cy[i]));
        zmc[n] = z;
    }
    __syncthreads();

    float q0, q1, n0, n1, den, w, rin;
    float* efb = ef + (size_t)b * FEAT_;
    float* mfb = mf + (size_t)b * FEAT_;
    float* cfb = cf + (size_t)b * FEAT_;

    // 1) bmm(e, A_em): logits E_em[i,n] = ex[i]*mx[n] + ey[i]*my[n]; vals e
    q0 = mx[n]; q1 = my[n]; n0 = n1 = den = 0.0f;
    for (int i = 0; i < M_; ++i) {
        w = __expf(fmaf(ex[i], q0, ey[i] * q1));
        n0 = fmaf(ex[i], w, n0); n1 = fmaf(ey[i], w, n1); den += w;
    }
    rin = 1.0f / den;
    efb[n] = n0 * rin; efb[M_ + n] = n1 * rin;

    // 2) bmm(e, A_ec): logits E_ec[i,n] = ex[i]*cx[n] + ey[i]*cy[n]; vals e
    q0 = cx[n]; q1 = cy[n]; n0 = n1 = den = 0.0f;
    for (int i = 0; i < M_; ++i) {
        w = __expf(fmaf(ex[i], q0, ey[i] * q1));
        n0 = fmaf(ex[i], w, n0); n1 = fmaf(ey[i], w, n1); den += w;
    }
    rin = 1.0f / den;
    efb[2 * M_ + n] = n0 * rin; efb[3 * M_ + n] = n1 * rin;

    // 3) bmm(m, A_me): logits E_em[n,i] = ex[n]*mx[i] + ey[n]*my[i]; vals m
    q0 = ex[n]; q1 = ey[n]; n0 = n1 = den = 0.0f;
    for (int i = 0; i < M_; ++i) {
        w = __expf(fmaf(mx[i], q0, my[i] * q1));
        n0 = fmaf(mx[i], w, n0); n1 = fmaf(my[i], w, n1); den += w;
    }
    rin = 1.0f / den;
    mfb[n] = n0 * rin; mfb[M_ + n] = n1 * rin;

    // 4) bmm(m, A_mc): w = exp(E_mc[i,n]) / Zrow_mc[i]; vals m (no per-n denom)
    q0 = cx[n]; q1 = cy[n]; n0 = n1 = 0.0f;
    for (int i = 0; i < M_; ++i) {
        w = __expf(fmaf(mx[i], q0, my[i] * q1)) / zmc[i];
        n0 = fmaf(mx[i], w, n0); n1 = fmaf(my[i], w, n1);
    }
    mfb[2 * M_ + n] = n0; mfb[3 * M_ + n] = n1;

    // 5) bmm(c, A_ce): logits E_ec[n,i] = ex[n]*cx[i] + ey[n]*cy[i]; vals c
    q0 = ex[n]; q1 = ey[n]; n0 = n1 = den = 0.0f;
    for (int i = 0; i < M_; ++i) {
        w = __expf(fmaf(cx[i], q0, cy[i] * q1));
        n0 = fmaf(cx[i], w, n0); n1 = fmaf(cy[i], w, n1); den += w;
    }
    rin = 1.0f / den;
    cfb[n] = n0 * rin; cfb[M_ + n] = n1 * rin;

    // 6) bmm(c, A_cm): logits E_mc[n,i] = mx[n]*cx[i] + my[n]*cy[i]; vals c
    q0 = mx[n]; q1 = my[n]; n0 = n1 = den = 0.0f;
    for (int i = 0; i < M_; ++i) {
        w = __expf(fmaf(cx[i], q0, cy[i] * q1));
        n0 = fmaf(cx[i], w, n0); n1 = fmaf(cy[i], w, n1); den += w;
    }
    rin = 1.0f / den;
    cfb[2 * M_ + n] = n0 * rin; cfb[3 * M_ + n] = n1 * rin;
}

// FC3 + sigmoid: one block per (sample, head); dot over CLS_=1024.
__global__ __launch_bounds__(256)
void head_kernel(const float* __restrict__ hid,
                 const float* W3e, const float* b3e,
                 const float* W3m, const float* b3m,
                 const float* W3c, const float* b3c,
                 float* __restrict__ out) {
    __shared__ float red[256];
    const int b = blockIdx.x, h = blockIdx.y, t = threadIdx.x;
    const float* W3   = (h == 0) ? W3e : (h == 1) ? W3m : W3c;
    const float  bias = ((h == 0) ? b3e : (h == 1) ? b3m : b3c)[0];
    const float* row  = hid + ((size_t)h * B_ + b) * CLS_;
    float s = 0.0f;
    for (int i = t; i < CLS_; i += 256) s = fmaf(row[i], W3[i], s);
    red[t] = s;
    __syncthreads();
    for (int st = 128; st > 0; st >>= 1) {
        if (t < st) red[t] += red[t + st];
        __syncthreads();
    }
    if (t == 0) out[h * B_ + b] = 1.0f / (1.0f + __expf(-(red[0] + bias)));
}

extern "C" void kernel_launch(void* const* d_in, const int* in_sizes, int n_in,
                              void* d_out, int out_size, void* d_ws, size_t ws_size,
                              hipStream_t stream) {
    const float* expr = (const float*)d_in[0];
    const float* mut  = (const float*)d_in[1];
    const float* cna  = (const float*)d_in[2];
    const float* Wex  = (const float*)d_in[3];
    const float* Wey  = (const float*)d_in[4];
    const float* Wmx  = (const float*)d_in[5];
    const float* Wmy  = (const float*)d_in[6];
    const float* Wcx  = (const float*)d_in[7];
    const float* Wcy  = (const float*)d_in[8];
    const float* W2e  = (const float*)d_in[9];
    const float* b2e  = (const float*)d_in[10];
    const float* W2m  = (const float*)d_in[11];
    const float* b2m  = (const float*)d_in[12];
    const float* W2c  = (const float*)d_in[13];
    const float* b2c  = (const float*)d_in[14];
    const float* W3e  = (const float*)d_in[15];
    const float* b3e  = (const float*)d_in[16];
    const float* W3m  = (const float*)d_in[17];
    const float* b3m  = (const float*)d_in[18];
    const float* W3c  = (const float*)d_in[19];
    const float* b3c  = (const float*)d_in[20];

    // Workspace layout (floats): fc1[6*B*M] | feat[3*B*FEAT] | hid[3*B*CLS]
    float* ws   = (float*)d_ws;
    float* fc1  = ws;                                   // 294912 f
    float* feat = fc1 + (size_t)6 * B_ * M_;            // 589824 f
    float* hid  = feat + (size_t)3 * B_ * FEAT_;        // 393216 f

    // ---- Stage 1: six FC1 GEMMs (memory-bound; fp32 WMMA) ----
    GemmBatch g1 = {};
    g1.A[0] = expr; g1.A[1] = expr;
    g1.A[2] = mut;  g1.A[3] = mut;
    g1.A[4] = cna;  g1.A[5] = cna;
    g1.W[0] = Wex;  g1.W[1] = Wey;
    g1.W[2] = Wmx;  g1.W[3] = Wmy;
    g1.W[4] = Wcx;  g1.W[5] = Wcy;
    for (int i = 0; i < 6; ++i) {
        g1.bias[i] = nullptr;
        g1.C[i] = fc1 + (size_t)i * B_ * M_;
    }
    wmma_gemm_f32<<<dim3(B_ / 16, M_ / 16, 6), 32, 0, stream>>>(g1, NG_, M_, 0);

    // ---- Stage 2: fused normalize + rank-2 softmax attention ----
    float* ef = feat;
    float* mf = feat + (size_t)B_ * FEAT_;
    float* cf = feat + (size_t)2 * B_ * FEAT_;
    attn_kernel<<<dim3(B_), M_, 0, stream>>>(fc1, ef, mf, cf);

    // ---- Stage 3: three FC2 GEMMs + bias + ReLU (fp32 WMMA) ----
    GemmBatch g2 = {};
    g2.A[0] = ef;  g2.A[1] = mf;  g2.A[2] = cf;
    g2.W[0] = W2e; g2.W[1] = W2m; g2.W[2] = W2c;
    g2.bias[0] = b2e; g2.bias[1] = b2m; g2.bias[2] = b2c;
    for (int i = 0; i < 3; ++i) g2.C[i] = hid + (size_t)i * B_ * CLS_;
    wmma_gemm_f32<<<dim3(B_ / 16, CLS_ / 16, 3), 32, 0, stream>>>(g2, FEAT_, CLS_, 1);

    // ---- Stage 4: FC3 + sigmoid heads ----
    head_kernel<<<dim3(B_, 3), 256, 0, stream>>>(hid, W3e, b3e, W3m, b3m, W3c, b3c,
                                                 (float*)d_out);
}